// KalmanFilter_10831907520857
// MI455X (gfx1250) — compile-verified
//
#include <hip/hip_runtime.h>
#include <hip/hip_bf16.h>
#include <math.h>

// ---------------------------------------------------------------------------
// Batched EKF: predict + nearest-8-of-N sensor selection + update.
// Distance tiles computed with V_WMMA_F32_16X16X4_F32:
//   d^2 = [px, py, px^2+py^2, 1] . [-2sx, -2sy, 1, sx^2+sy^2]
// 4-tile software pipelining: 4 clustered global_load_b64 -> 4 WMMAs ->
// compare/insert phase, to hide L2 latency at low occupancy.
// ---------------------------------------------------------------------------

typedef __attribute__((ext_vector_type(2))) float v2f;
typedef __attribute__((ext_vector_type(8))) float v8f;

#define CH 16          // sensor chunks (kernel-1 grid.y); workspace = B*CH*8*(4+4) bytes
#define UNR 4          // sensor tiles per inner iteration
#define MAPSZ 100.0f
#define BIGF  3.0e38f

// ============================ Kernel 1 =====================================
__global__ __launch_bounds__(32)
void ekf_topk_kernel(const float* __restrict__ x0,
                     const float* __restrict__ delta,
                     const float* __restrict__ sensors,
                     float* __restrict__ wsd,
                     int* __restrict__ wsi,
                     int B, int N)
{
    const int lane = threadIdx.x;        // 0..31
    const int col  = lane & 15;          // sensor column / filter row (low half)
    const int hi   = lane >> 4;          // upper half-wave flag

    const int fbase = blockIdx.x * 16;
    const int chunk = blockIdx.y;
    const int chunk_len = (N + CH - 1) / CH;
    const int start = chunk * chunk_len;
    const int end   = (start + chunk_len < N) ? (start + chunk_len) : N;
    const int tiles = (end > start) ? ((end - start + 15) >> 4) : 0;

    // ---- predicted, clipped position of filter (fbase+col) ----
    const int f = fbase + col;
    float px = 0.0f, py = 0.0f;
    if (f < B) {
        px = x0[f * 3 + 0] + delta[f * 3 + 0];
        py = x0[f * 3 + 1] + delta[f * 3 + 1];
        px = fminf(fmaxf(px, 0.0f), MAPSZ);
        py = fminf(fmaxf(py, 0.0f), MAPSZ);
    }
    const float pp = px * px + py * py;

    // A matrix 16x4 f32 (2 VGPRs): VGPR0 = {K0 | K2}, VGPR1 = {K1 | K3}
    v2f a;
    a.x = hi ? pp   : px;    // K2 : K0
    a.y = hi ? 1.0f : py;    // K3 : K1

    // per-lane top-8 lists, one per C VGPR (filter row v + 8*hi)
    float bd[8][8];
    int   bi[8][8];
    float bmax[8];
    int   bslot[8];
#pragma unroll
    for (int v = 0; v < 8; ++v) {
#pragma unroll
        for (int s = 0; s < 8; ++s) { bd[v][s] = BIGF; bi[v][s] = 0; }
        bmax[v] = BIGF; bslot[v] = 0;
    }

    const int iters = (tiles + UNR - 1) / UNR;
    for (int it = 0; it < iters; ++it) {
        const int tb = it * UNR;

        int   jj[UNR];
        float sx[UNR], sy[UNR];
        // clustered, unconditional loads (address clamped; invalid handled below)
#pragma unroll
        for (int u = 0; u < UNR; ++u) {
            jj[u] = start + ((tb + u) << 4) + col;
            int jc = jj[u] < (N - 1) ? jj[u] : (N - 1);
            jc = jc < 0 ? 0 : jc;
            const v2f sv = *(const v2f*)(sensors + 2 * jc);
            sx[u] = sv.x;
            sy[u] = sv.y;
        }

        v8f c[UNR];
#pragma unroll
        for (int u = 0; u < UNR; ++u) {
            const bool valid = (jj[u] < end);
            const float ss = valid ? (sx[u] * sx[u] + sy[u] * sy[u]) : 1.0e30f;
            const float k2 = valid ? 1.0f : 0.0f;
            // B matrix 4x16 f32 (2 VGPRs): VGPR0 = {K0 | K2}, VGPR1 = {K1 | K3}
            v2f b;
            b.x = hi ? k2 : (-2.0f * sx[u]);
            b.y = hi ? ss : (-2.0f * sy[u]);
            v8f cz = {0.f, 0.f, 0.f, 0.f, 0.f, 0.f, 0.f, 0.f};
            c[u] = __builtin_amdgcn_wmma_f32_16x16x4_f32(
                       /*neg_a=*/false, a, /*neg_b=*/false, b,
                       /*c_mod=*/(short)0, cz, /*reuse_a=*/false, /*reuse_b=*/false);
        }

        // c[u][v] is d^2 of (filter row v+8*hi, sensor jj[u]) -> maintain top-8
#pragma unroll
        for (int u = 0; u < UNR; ++u) {
#pragma unroll
            for (int v = 0; v < 8; ++v) {
                const float d2 = c[u][v];
                if (d2 < bmax[v]) {
                    const int sl = bslot[v];
#pragma unroll
                    for (int s = 0; s < 8; ++s)
                        if (s == sl) { bd[v][s] = d2; bi[v][s] = jj[u]; }
                    float m = bd[v][0]; int ms = 0;
#pragma unroll
                    for (int s = 1; s < 8; ++s)
                        if (bd[v][s] > m) { m = bd[v][s]; ms = s; }
                    bmax[v] = m; bslot[v] = ms;
                }
            }
        }
    }

    // ---- cross-lane merge via LDS: row r candidates live in 16 lanes x 8 ----
    __shared__ float sdl[8][32][8];
    __shared__ int   sil[8][32][8];
#pragma unroll
    for (int v = 0; v < 8; ++v)
#pragma unroll
        for (int s = 0; s < 8; ++s) {
            sdl[v][lane][s] = bd[v][s];
            sil[v][lane][s] = bi[v][s];
        }
    __syncthreads();

    if (lane < 16) {
        const int row = lane;                 // filter row 0..15
        const int v   = row & 7;
        const int lo  = (row >= 8) ? 16 : 0;
        const int fo  = fbase + row;
        if (fo < B) {
            const long long base = ((long long)fo * CH + chunk) * 8;
            // 8 ascending lexicographic-(d2,idx) min passes over 128 candidates
            float pd = -1.0f; int pi = -2147483647;
            for (int m = 0; m < 8; ++m) {
                float best = BIGF; int bidx = 2147483647;
                for (int k = 0; k < 16; ++k) {
#pragma unroll
                    for (int s = 0; s < 8; ++s) {
                        const float dj = sdl[v][lo + k][s];
                        const int   ij = sil[v][lo + k][s];
                        const bool gt = (dj > pd)   || (dj == pd   && ij > pi);
                        const bool lt = (dj < best) || (dj == best && ij < bidx);
                        if (gt && lt) { best = dj; bidx = ij; }
                    }
                }
                wsd[base + m] = best;
                wsi[base + m] = bidx;
                pd = best; pi = bidx;
            }
        }
    }
}

// ============================ Kernel 2 =====================================
// One thread per filter: predict P, merge CH*8 candidates -> global top-8
// (ascending d, ties by index), exact H/innovation, Gauss-Jordan 8x8 solve,
// Kalman update. Writes [x_new(3) | P_new(9)] per filter.
__global__ __launch_bounds__(256)
void ekf_update_kernel(const float* __restrict__ x0,
                       const float* __restrict__ P0,
                       const float* __restrict__ sensors,
                       const float* __restrict__ delta,
                       const float* __restrict__ z,
                       const float* __restrict__ wsd,
                       const int* __restrict__ wsi,
                       float* __restrict__ out,
                       int B, int N)
{
    const int f = blockIdx.x * blockDim.x + threadIdx.x;
    if (f >= B) return;

    // ---- predict ----
    const float X  = x0[f * 3 + 0] + delta[f * 3 + 0];
    const float Y  = x0[f * 3 + 1] + delta[f * 3 + 1];
    const float TH = x0[f * 3 + 2] + delta[f * 3 + 2];
    const float px = fminf(fmaxf(X, 0.0f), MAPSZ);
    const float py = fminf(fmaxf(Y, 0.0f), MAPSZ);
    const float sn = sinf(TH), cs = cosf(TH);

    float P[3][3];
#pragma unroll
    for (int i = 0; i < 3; ++i)
#pragma unroll
        for (int j = 0; j < 3; ++j)
            P[i][j] = P0[f * 9 + i * 3 + j];

    const float fa = -0.2f * sn;   // F[0][2]
    const float fb =  0.2f * cs;   // F[1][2]

    float T[3][3], Pp[3][3];
#pragma unroll
    for (int j = 0; j < 3; ++j) {
        T[0][j] = P[0][j] + fa * P[2][j];
        T[1][j] = P[1][j] + fb * P[2][j];
        T[2][j] = P[2][j];
    }
#pragma unroll
    for (int i = 0; i < 3; ++i) {
        Pp[i][0] = T[i][0] + fa * T[i][2];
        Pp[i][1] = T[i][1] + fb * T[i][2];
        Pp[i][2] = T[i][2];
    }
    // process noise Q
    const float TPN  = 6.2831853071795864f * 0.01f;          // 2*pi*THETA_NOISE
    const float TN2  = TPN * TPN;
    const float SN23 = (0.02f * 0.02f) / 3.0f;               // SPEED_NOISE^2/3
    const float sx2  = SN23 * cs * cs + 0.2f * TN2 * sn * sn;
    const float sy2  = SN23 * sn * sn + 0.2f * TN2 * cs * cs;
    const float sxy  = (SN23 * 0.5f + (0.2f * TPN) * (0.2f * TPN)) * sinf(2.0f * TH);
    const float sxp  = -0.2f * TN2 * sn;
    const float syp  =  0.2f * TN2 * cs;
    Pp[0][0] += sx2; Pp[0][1] += sxy; Pp[0][2] += sxp;
    Pp[1][0] += sxy; Pp[1][1] += sy2; Pp[1][2] += syp;
    Pp[2][0] += sxp; Pp[2][1] += syp; Pp[2][2] += TN2;

    // ---- merge per-chunk candidates: global ascending top-8 ----
    const long long cbase = (long long)f * (CH * 8);
    float dme[8]; int ime[8];
    {
        float pd = -1.0f; int pi = -2147483647;
        for (int m = 0; m < 8; ++m) {
            float best = BIGF; int bidx = 2147483647;
            for (int j = 0; j < CH * 8; ++j) {
                const float dj = wsd[cbase + j];
                const int   ij = wsi[cbase + j];
                const bool gt = (dj > pd)   || (dj == pd   && ij > pi);
                const bool lt = (dj < best) || (dj == best && ij < bidx);
                if (gt && lt) { best = dj; bidx = ij; }
            }
            dme[m] = best; ime[m] = bidx;
            pd = best; pi = bidx;
        }
    }

    // ---- build H, innovation (exact distances from coordinates) ----
    float H0[8], H1[8], yv[8];
#pragma unroll
    for (int m = 0; m < 8; ++m) {
        int idx = ime[m];
        idx = (idx >= 0 && idx < N) ? idx : 0;   // safety for degenerate cases
        const float sx = sensors[2 * idx];
        const float sy = sensors[2 * idx + 1];
        const float dx = px - sx;
        const float dy = py - sy;
        const float d  = sqrtf(dx * dx + dy * dy);
        H0[m] = dx / d;
        H1[m] = dy / d;
        yv[m] = z[f * 8 + m] - d;
    }

    // ---- S = H P H^T + R ; solve S*[W | w] = [ (P H^T)^T | y ] ----
    float PH[3][8];
#pragma unroll
    for (int m = 0; m < 8; ++m) {
        PH[0][m] = Pp[0][0] * H0[m] + Pp[0][1] * H1[m];
        PH[1][m] = Pp[1][0] * H0[m] + Pp[1][1] * H1[m];
        PH[2][m] = Pp[2][0] * H0[m] + Pp[2][1] * H1[m];
    }
    float S[8][8];
#pragma unroll
    for (int m = 0; m < 8; ++m)
#pragma unroll
        for (int n = 0; n < 8; ++n)
            S[m][n] = H0[m] * PH[0][n] + H1[m] * PH[1][n] + ((m == n) ? 0.09f : 0.0f);

    float Rh[8][4];
#pragma unroll
    for (int m = 0; m < 8; ++m) {
        Rh[m][0] = PH[0][m];
        Rh[m][1] = PH[1][m];
        Rh[m][2] = PH[2][m];
        Rh[m][3] = yv[m];
    }

    // Gauss-Jordan, fully unrolled (S is SPD + 0.09*I -> no pivoting needed)
#pragma unroll
    for (int k = 0; k < 8; ++k) {
        const float inv = 1.0f / S[k][k];
#pragma unroll
        for (int j = 0; j < 8; ++j) S[k][j] *= inv;
#pragma unroll
        for (int c = 0; c < 4; ++c) Rh[k][c] *= inv;
#pragma unroll
        for (int i = 0; i < 8; ++i) {
            if (i == k) continue;
            const float fac = S[i][k];
#pragma unroll
            for (int j = 0; j < 8; ++j) S[i][j] -= fac * S[k][j];
#pragma unroll
            for (int c = 0; c < 4; ++c) Rh[i][c] -= fac * Rh[k][c];
        }
    }

    // ---- update: x' = x + PH^T * w ; P' = P - PH^T * W ----
    float xn0 = px, xn1 = py, xn2 = TH;
    float Pn[3][3];
#pragma unroll
    for (int i = 0; i < 3; ++i)
#pragma unroll
        for (int j = 0; j < 3; ++j)
            Pn[i][j] = Pp[i][j];
#pragma unroll
    for (int m = 0; m < 8; ++m) {
        xn0 += PH[0][m] * Rh[m][3];
        xn1 += PH[1][m] * Rh[m][3];
        xn2 += PH[2][m] * Rh[m][3];
#pragma unroll
        for (int i = 0; i < 3; ++i)
#pragma unroll
            for (int j = 0; j < 3; ++j)
                Pn[i][j] -= PH[i][m] * Rh[m][j];
    }

    float* o = out + (long long)f * 12;
    o[0] = xn0; o[1] = xn1; o[2] = xn2;
#pragma unroll
    for (int i = 0; i < 3; ++i)
#pragma unroll
        for (int j = 0; j < 3; ++j)
            o[3 + i * 3 + j] = Pn[i][j];
}

// ============================ Launcher =====================================
extern "C" void kernel_launch(void* const* d_in, const int* in_sizes, int n_in,
                              void* d_out, int out_size, void* d_ws, size_t ws_size,
                              hipStream_t stream)
{
    const float* x0      = (const float*)d_in[0];   // [B,3]
    const float* P0      = (const float*)d_in[1];   // [B,3,3]
    const float* sensors = (const float*)d_in[2];   // [N,2]
    const float* delta   = (const float*)d_in[3];   // [B,3]
    const float* z       = (const float*)d_in[4];   // [B,8]

    const int B = in_sizes[0] / 3;
    const int N = in_sizes[2] / 2;

    float* wsd = (float*)d_ws;                       // B*CH*8 floats
    int*   wsi = (int*)(wsd + (size_t)B * CH * 8);   // B*CH*8 ints

    dim3 g1((B + 15) / 16, CH);
    ekf_topk_kernel<<<g1, 32, 0, stream>>>(x0, delta, sensors, wsd, wsi, B, N);

    const int t2 = 256;
    ekf_update_kernel<<<(B + t2 - 1) / t2, t2, 0, stream>>>(
        x0, P0, sensors, delta, z, wsd, wsi, (float*)d_out, B, N);
}